// BaseRNN_26456998543907
// MI455X (gfx1250) — compile-verified
//
#include <hip/hip_runtime.h>
#include <hip/hip_bf16.h>

typedef float v2f __attribute__((ext_vector_type(2)));
typedef float v8f __attribute__((ext_vector_type(8)));

namespace {
constexpr int S_LEN = 512;
constexpr int B_DIM = 16384;
constexpr int A_DIM = 2;
constexpr int BA    = B_DIM * A_DIM;        // 32768 elements
constexpr int HID   = 16;
constexpr int EMB   = 8;
constexpr int NPART = 1000;
constexpr int PROJ_N = NPART * HID;         // 16000 floats = 64 KB LUT

constexpr int WAVE = 32;
constexpr int WAVES_PER_BLOCK = 8;
constexpr int BLOCK_THREADS = WAVES_PER_BLOCK * WAVE;             // 256
constexpr int ELEMS_PER_WAVE = 16;                                // one WMMA tile
constexpr int ELEMS_PER_BLOCK = WAVES_PER_BLOCK * ELEMS_PER_WAVE; // 128
constexpr int GRID = BA / ELEMS_PER_BLOCK;                        // 256 blocks
}

__device__ __forceinline__ float fast_tanh(float x) {
#if __has_builtin(__builtin_amdgcn_tanhf)
  return __builtin_amdgcn_tanhf(x);       // v_tanh_f32 (gfx1250 trans op)
#else
  float e = __builtin_amdgcn_exp2f(x * 2.8853900817779268f); // 2*log2(e)
  return 1.0f - 2.0f * __builtin_amdgcn_rcpf(e + 1.0f);
#endif
}

__device__ __forceinline__ float fast_sigmoid(float x) {
  return __builtin_amdgcn_rcpf(1.0f + __builtin_amdgcn_exp2f(-x * 1.4426950408889634f));
}

// Kernel 1: fold embedding + b1 into a 1000x16 LUT:
//   proj[p][j] = b1[j] + sum_k emb[p][k] * W1[2+k][j]
__global__ void build_emb_proj(const float* __restrict__ emb,
                               const float* __restrict__ W1,
                               const float* __restrict__ b1,
                               float* __restrict__ proj) {
  int t = blockIdx.x * blockDim.x + threadIdx.x;
  if (t >= PROJ_N) return;
  int p = t >> 4;
  int j = t & 15;
  float acc = b1[j];
#pragma unroll
  for (int k = 0; k < EMB; ++k)
    acc = fmaf(emb[p * EMB + k], W1[(2 + k) * HID + j], acc);
  proj[t] = acc;
}

// Kernel 2: sequential scan, one wave owns 16 elements. Per step one
// V_WMMA_F32_16X16X4_F32 computes all 16x16 hidden pre-activations:
//   D[j,i] = W1[0][j]*state_i + W1[1][j]*r_i + LUT[pid_i][j]
// Software-pipelined one step deep: global loads for s+1 are issued at the
// top of iteration s, and the LDS C-gather for s+1 at the bottom, so the
// WMMA critical path never waits on HBM.
__launch_bounds__(BLOCK_THREADS)
__global__ void rnn_scan(const int*   __restrict__ act,
                         const float* __restrict__ rew,
                         const int*   __restrict__ pid,
                         const float* __restrict__ proj,
                         const float* __restrict__ W1,
                         const float* __restrict__ W2,
                         const float* __restrict__ b2v,
                         float* __restrict__ out) {
  __shared__ __align__(16) float s_proj[PROJ_N];   // 64 KB LUT in LDS
  for (int i = threadIdx.x; i < PROJ_N; i += BLOCK_THREADS)
    s_proj[i] = proj[i];
  __syncthreads();

  const int  lane = threadIdx.x & (WAVE - 1);
  const int  wid  = threadIdx.x >> 5;
  const int  i    = lane & 15;                 // element index within wave tile
  const bool hi   = lane >= 16;
  const int  elem = (blockIdx.x * WAVES_PER_BLOCK + wid) * ELEMS_PER_WAVE + i;
  const int  b    = elem >> 1;                 // A_DIM == 2
  const int  coff = i * HID + (hi ? 8 : 0);    // unused; kept pattern simple

  // A-matrix (constant): M = lane%16 = hidden j; VGPR0 <- K=0|2, VGPR1 <- K=1|3.
  // Upper-half K columns are zero so B's upper-half payload is don't-care.
  const int j = i;
  v2f amat;
  amat.x = hi ? 0.0f : W1[j];          // K0: state coeff   (K2 = 0)
  amat.y = hi ? 0.0f : W1[HID + j];    // K1: reward coeff  (K3 = 0)

  // W2 slice this lane reduces over (hidden rows g+8*hi)
  float w2h[8];
#pragma unroll
  for (int g = 0; g < 8; ++g) w2h[g] = W2[g + (hi ? 8 : 0)];
  const float bias2 = b2v[0];

  // ---- pipeline prologue: step 0 operands ----
  float r_cur = __builtin_nontemporal_load(&rew[elem]);
  int   a_cur = __builtin_nontemporal_load(&act[elem]);
  int   p_cur = pid[b];
  float4 c0 = *(const float4*)&s_proj[p_cur * HID + (hi ? 8 : 0)];
  float4 c1 = *(const float4*)&s_proj[p_cur * HID + (hi ? 8 : 0) + 4];

  float state = 0.0f;

  for (int s = 0; s < S_LEN; ++s) {
    const int idx = s * BA + elem;

    // issue next step's global loads now (wrap keeps addresses valid)
    const int sn   = (s + 1) & (S_LEN - 1);
    const int idxn = sn * BA + elem;
    float r_nxt = __builtin_nontemporal_load(&rew[idxn]);
    int   a_nxt = __builtin_nontemporal_load(&act[idxn]);
    int   p_nxt = pid[sn * B_DIM + b];

    // stream-ahead prefetch two steps out (uniform clamp, SALU select)
    const int spf = (s + 2 < S_LEN) ? s + 2 : S_LEN - 1;
    __builtin_prefetch(&rew[spf * BA + elem], 0, 1);
    __builtin_prefetch(&act[spf * BA + elem], 0, 1);
    __builtin_prefetch(&pid[spf * B_DIM + b], 0, 1);

    // D = A x B + C  -> 16 hidden pre-activations for 16 elements
    v8f cmat;
    cmat[0] = c0.x; cmat[1] = c0.y; cmat[2] = c0.z; cmat[3] = c0.w;
    cmat[4] = c1.x; cmat[5] = c1.y; cmat[6] = c1.z; cmat[7] = c1.w;
    v2f bmat;
    bmat.x = state;
    bmat.y = r_cur;
    v8f d = __builtin_amdgcn_wmma_f32_16x16x4_f32(
        /*neg_a=*/false, amat, /*neg_b=*/false, bmat,
        /*c_mod=*/(short)0, cmat, /*reuse_a=*/false, /*reuse_b=*/false);

    // h = relu(tanh(pre)); partial dot with W2 over this lane's 8 hidden rows
    float partial = 0.0f;
#pragma unroll
    for (int g = 0; g < 8; ++g) {
      float h = fmaxf(fast_tanh(d[g]), 0.0f);
      partial = fmaf(h, w2h[g], partial);
    }
    // complete the 16-wide hidden reduction across wave halves
    float other = __shfl_xor(partial, 16, 32);
    float mask  = (a_cur == 1) ? 1.0f : 0.0f;
    float upd   = (partial + other + bias2) * mask;

    state = fast_sigmoid(state + upd);

    // both halves hold identical state: unpredicated store (same addr, same
    // data from lane l and l+16) avoids per-iteration EXEC save/restore
    __builtin_nontemporal_store(state, &out[idx]);

    // LDS C-gather for next step; pid load latency is covered by this
    // iteration's compute
    const float4* cp = (const float4*)&s_proj[p_nxt * HID + (hi ? 8 : 0)];
    c0 = cp[0];
    c1 = cp[1];

    r_cur = r_nxt;
    a_cur = a_nxt;
    p_cur = p_nxt;
    (void)coff;
  }
}

extern "C" void kernel_launch(void* const* d_in, const int* in_sizes, int n_in,
                              void* d_out, int out_size, void* d_ws, size_t ws_size,
                              hipStream_t stream) {
  const int*   act = (const int*)d_in[0];    // c_Action (S,B,A) int32
  const float* rew = (const float*)d_in[1];  // c_Reward (S,B,A) f32
  const int*   pid = (const int*)d_in[2];    // c_ParticipantID (S,B,1) int32
  const float* emb = (const float*)d_in[3];  // emb_table (1000,8)
  const float* W1  = (const float*)d_in[4];  // (10,16) row-major
  const float* b1  = (const float*)d_in[5];  // (16,)
  const float* W2  = (const float*)d_in[6];  // (16,1)
  const float* b2  = (const float*)d_in[7];  // (1,)
  float* out  = (float*)d_out;               // (S,B,A) f32
  float* proj = (float*)d_ws;                // 16000 floats scratch LUT

  build_emb_proj<<<(PROJ_N + 255) / 256, 256, 0, stream>>>(emb, W1, b1, proj);
  rnn_scan<<<GRID, BLOCK_THREADS, 0, stream>>>(act, rew, pid, proj, W1, W2, b2, out);
}